// Router_29738353557792
// MI455X (gfx1250) — compile-verified
//
#include <hip/hip_runtime.h>
#include <hip/hip_bf16.h>
#include <math.h>

// CDNA5 (gfx1250) router kernel: fp32 GEMM via V_WMMA_F32_16X16X4_F32,
// LDS-staged W tiles, exact top-8 + masked softmax per token.

typedef __attribute__((ext_vector_type(2))) float v2f;
typedef __attribute__((ext_vector_type(4))) float v4f;
typedef __attribute__((ext_vector_type(8))) float v8f;

#define EMBD    4096
#define NEXP    64
#define TOPK    8
#define KCHUNK  128          // K elements of W staged in LDS per block iteration
#define WPITCH  132          // floats per expert row in LDS (132 % 64 == 4 -> conflict-free b64 reads)
#define LPITCH  68           // floats per token row for logits staging (68 % 64 == 4)
#define WAVES   8            // waves per block (256 threads, wave32)

enum {
  SW_FLOATS   = NEXP * WPITCH,          // 8448
  LG_FLOATS   = WAVES * 16 * LPITCH,    // 8704
  SMEM_FLOATS = (SW_FLOATS > LG_FLOATS ? SW_FLOATS : LG_FLOATS)
};

__global__ __launch_bounds__(256) void router_wmma_f32_kernel(
    const float* __restrict__ x,      // [ntok, EMBD]
    const float* __restrict__ W,      // [NEXP, EMBD]
    float* __restrict__ probs,        // [ntok, NEXP]
    float* __restrict__ idx_out,      // [ntok, TOPK] (indices as floats)
    int ntiles)                        // ntok / 16
{
  __shared__ float smem[SMEM_FLOATS];

  const int tid  = threadIdx.x;
  const int lane = tid & 31;
  const int wave = tid >> 5;
  const int half = lane >> 4;     // 0: K=0..1 / M=j ; 1: K=2..3 / M=j+8
  const int nl   = lane & 15;     // row (A) / column (B,C) index within tile

  int tile = blockIdx.x * WAVES + wave;
  if (tile >= ntiles) tile = ntiles - 1;   // benign duplicate work on ragged grids
  const long tok0 = (long)tile * 16;

  // A fragment source: lane (half,nl) loads x[tok0+nl][k + 2*half .. +1] each step.
  const float* aBase = x + (tok0 + nl) * (long)EMBD + 2 * half;

  v8f acc0 = {}, acc1 = {}, acc2 = {}, acc3 = {};

  for (int k0 = 0; k0 < EMBD; k0 += KCHUNK) {
    __syncthreads();   // previous chunk fully consumed before overwrite

    // ---- cooperative stage of W[:, k0 : k0+KCHUNK] into LDS ----
    {
      const int e   = tid >> 2;        // 0..63
      const int sub = tid & 3;         // 4 threads per expert row
      const float* wp = W + (long)e * EMBD + k0 + sub * 4;
      float* sp = &smem[e * WPITCH + sub * 4];
#pragma unroll
      for (int i = 0; i < 8; ++i) {
        v4f w = *(const v4f*)(wp + i * 16);
        *(v4f*)(sp + i * 16) = w;
      }
    }
    __syncthreads();

    // ---- WMMA K-loop over the staged chunk ----
    const float* ap = aBase + k0;
#pragma unroll 4
    for (int kc = 0; kc < KCHUNK; kc += 4) {
      v2f a = *(const v2f*)(ap + kc);
      const float* bl = &smem[nl * WPITCH + kc + 2 * half];
      v2f b0 = *(const v2f*)(bl + 0  * 16 * WPITCH);
      v2f b1 = *(const v2f*)(bl + 1  * 16 * WPITCH);
      v2f b2 = *(const v2f*)(bl + 2  * 16 * WPITCH);
      v2f b3 = *(const v2f*)(bl + 3  * 16 * WPITCH);
      acc0 = __builtin_amdgcn_wmma_f32_16x16x4_f32(false, a, false, b0, (short)0, acc0, false, false);
      acc1 = __builtin_amdgcn_wmma_f32_16x16x4_f32(false, a, false, b1, (short)0, acc1, false, false);
      acc2 = __builtin_amdgcn_wmma_f32_16x16x4_f32(false, a, false, b2, (short)0, acc2, false, false);
      acc3 = __builtin_amdgcn_wmma_f32_16x16x4_f32(false, a, false, b3, (short)0, acc3, false, false);
    }
  }

  // ---- spill logits tile to LDS (C layout: VGPR j -> M = j + 8*half, N = nl) ----
  __syncthreads();
  float* lrow = &smem[wave * 16 * LPITCH];
#pragma unroll
  for (int j = 0; j < 8; ++j) {
    const int m = j + 8 * half;
    lrow[m * LPITCH +  0 + nl] = acc0[j];
    lrow[m * LPITCH + 16 + nl] = acc1[j];
    lrow[m * LPITCH + 32 + nl] = acc2[j];
    lrow[m * LPITCH + 48 + nl] = acc3[j];
  }
  __syncthreads();

  // ---- per-token top-8 + masked softmax (lanes 0..15, one token each) ----
  if (lane < 16) {
    const int m = lane;
    const long token = tok0 + m;

    float v[NEXP];
#pragma unroll
    for (int e = 0; e < NEXP; ++e) v[e] = lrow[m * LPITCH + e];

    float kept[TOPK];
    int   kidx[TOPK];
    float prevV = 3.0e38f;
    int   prevI = -1;
#pragma unroll
    for (int kk = 0; kk < TOPK; ++kk) {
      float best = -3.0e38f;
      int   bi   = 0;
      // eligible = strictly after (prevV, prevI) in (value desc, index asc) order;
      // strict '>' scan keeps the smallest index on ties (matches lax.top_k).
#pragma unroll
      for (int e = 0; e < NEXP; ++e) {
        const bool elig = (v[e] < prevV) || (v[e] == prevV && e > prevI);
        if (elig && v[e] > best) { best = v[e]; bi = e; }
      }
      kept[kk] = best; kidx[kk] = bi;
      prevV = best;    prevI = bi;
    }

    const float maxv = kept[0];
    float pk[TOPK];
    float sum = 0.0f;
#pragma unroll
    for (int kk = 0; kk < TOPK; ++kk) { pk[kk] = expf(kept[kk] - maxv); sum += pk[kk]; }
    const float inv = 1.0f / sum;

    float* po = probs + token * NEXP;
#pragma unroll 8
    for (int e = 0; e < NEXP; ++e) {
      float pv = 0.0f;   // exp(NEG_INF - max) == 0 exactly in fp32
#pragma unroll
      for (int kk = 0; kk < TOPK; ++kk)
        if (kidx[kk] == e) pv = pk[kk] * inv;
      po[e] = pv;
    }

    float* io = idx_out + token * TOPK;
#pragma unroll
    for (int kk = 0; kk < TOPK; ++kk) io[kk] = (float)kidx[kk];
  }
}

extern "C" void kernel_launch(void* const* d_in, const int* in_sizes, int n_in,
                              void* d_out, int out_size, void* d_ws, size_t ws_size,
                              hipStream_t stream) {
  const float* x = (const float*)d_in[0];   // [b*s, EMBD] fp32
  const float* W = (const float*)d_in[1];   // [NEXP, EMBD] fp32
  float* out = (float*)d_out;

  const int ntok   = in_sizes[0] / EMBD;    // 16384
  const int ntiles = ntok / 16;             // 1024
  float* probs = out;                        // [ntok, NEXP]
  float* idxo  = out + (long)ntok * NEXP;    // [ntok, TOPK]

  const int blocks = (ntiles + WAVES - 1) / WAVES;  // 128
  router_wmma_f32_kernel<<<blocks, 256, 0, stream>>>(x, W, probs, idxo, ntiles);
}